// _ConvBlock2_43018392436806
// MI455X (gfx1250) — compile-verified
//
#include <hip/hip_runtime.h>

typedef __attribute__((ext_vector_type(2))) float v2f;
typedef __attribute__((ext_vector_type(8))) float v8f;

#define C_DIM 128

// ---------------------------------------------------------------------------
// Weighted scatter-add: out[dst[e], :] += w[e] * src_feat[src[e], :]
// One thread handles one (edge, 4-channel) chunk -> float4 load + 4 f32 atomics.
// ---------------------------------------------------------------------------
__global__ void scatter_wadd4(const float* __restrict__ src_feat,
                              const int* __restrict__ sidx,
                              const int* __restrict__ didx,
                              const float* __restrict__ w,
                              float* __restrict__ out,
                              long long n_edge) {
    long long t = (long long)blockIdx.x * blockDim.x + threadIdx.x;
    long long total = n_edge * (C_DIM / 4);
    if (t >= total) return;
    long long e = t >> 5;              // / (128/4)
    int c4 = (int)(t & 31) * 4;
    int s = sidx[e];
    int d = didx[e];
    float we = w[e];
    const float4* sp = (const float4*)(src_feat + (size_t)s * C_DIM + c4);
    float4 v = *sp;
    float* op = out + (size_t)d * C_DIM + c4;
    atomicAdd(op + 0, we * v.x);
    atomicAdd(op + 1, we * v.y);
    atomicAdd(op + 2, we * v.z);
    atomicAdd(op + 3, we * v.w);
}

// ---------------------------------------------------------------------------
// One wave computes a 32x16 output strip (two 16x16 M-tiles sharing B frags):
//   Ynb   = H @ Wnb
//   Yroot = H @ Wroot + bias
// GUARD=false: rows m0..m0+31 all valid -> zero control flow in inner loop.
// GUARD=true : branchless clamp + select (v_cndmask), EXEC stays all-1s.
// ---------------------------------------------------------------------------
template <bool GUARD>
__device__ __forceinline__ void gemm_strip32(const float* __restrict__ H,
                                             const float* __restrict__ Wnb,
                                             const float* __restrict__ Wroot,
                                             const float* __restrict__ bias,
                                             float* __restrict__ Ynb,
                                             float* __restrict__ Yroot,
                                             int nRows, int m0, int n0, int lane) {
    const int half = lane >> 4;                  // 0 or 1
    const int l15  = lane & 15;
    const int col  = n0 + l15;

    const int row0 = m0 + l15;
    const int row1 = m0 + 16 + l15;
    const bool ok0 = !GUARD || (row0 < nRows);
    const bool ok1 = !GUARD || (row1 < nRows);
    const int sr0 = GUARD ? (ok0 ? row0 : 0) : row0;   // always-valid address
    const int sr1 = GUARD ? (ok1 ? row1 : 0) : row1;
    const float* __restrict__ h0 = H + (size_t)sr0 * C_DIM;
    const float* __restrict__ h1 = H + (size_t)sr1 * C_DIM;

    v8f acc_nb0 = {}, acc_nb1 = {}, acc_rt0 = {}, acc_rt1 = {};

    #pragma unroll 4
    for (int k = 0; k < C_DIM; k += 4) {
        const int ka = k + 2 * half;             // this lane covers K = ka, ka+1
        v2f a0 = *(const v2f*)(h0 + ka);         // one b64 load
        v2f a1 = *(const v2f*)(h1 + ka);
        if (GUARD) {
            a0.x = ok0 ? a0.x : 0.0f;  a0.y = ok0 ? a0.y : 0.0f;
            a1.x = ok1 ? a1.x : 0.0f;  a1.y = ok1 ? a1.y : 0.0f;
        }
        v2f bnb, brt;
        bnb.x = Wnb[(size_t)ka * C_DIM + col];
        bnb.y = Wnb[(size_t)(ka + 1) * C_DIM + col];
        brt.x = Wroot[(size_t)ka * C_DIM + col];
        brt.y = Wroot[(size_t)(ka + 1) * C_DIM + col];
        acc_nb0 = __builtin_amdgcn_wmma_f32_16x16x4_f32(false, a0, false, bnb,
                                                        (short)0, acc_nb0, false, false);
        acc_nb1 = __builtin_amdgcn_wmma_f32_16x16x4_f32(false, a1, false, bnb,
                                                        (short)0, acc_nb1, false, false);
        acc_rt0 = __builtin_amdgcn_wmma_f32_16x16x4_f32(false, a0, false, brt,
                                                        (short)0, acc_rt0, false, false);
        acc_rt1 = __builtin_amdgcn_wmma_f32_16x16x4_f32(false, a1, false, brt,
                                                        (short)0, acc_rt1, false, false);
    }

    const float bv = bias[col];
    #pragma unroll
    for (int r = 0; r < 8; ++r) {
        const int rr0 = m0 + r + 8 * half;       // C/D VGPR layout
        const int rr1 = rr0 + 16;
        if (!GUARD || (rr0 < nRows)) {
            Ynb[(size_t)rr0 * C_DIM + col]   = acc_nb0[r];
            Yroot[(size_t)rr0 * C_DIM + col] = acc_rt0[r] + bv;
        }
        if (!GUARD || (rr1 < nRows)) {
            Ynb[(size_t)rr1 * C_DIM + col]   = acc_nb1[r];
            Yroot[(size_t)rr1 * C_DIM + col] = acc_rt1[r] + bv;
        }
    }
}

__global__ void gemm_dual_wmma(const float* __restrict__ H,
                               const float* __restrict__ Wnb,
                               const float* __restrict__ Wroot,
                               const float* __restrict__ bias,
                               float* __restrict__ Ynb,
                               float* __restrict__ Yroot,
                               int nRows) {
    const int lane = threadIdx.x & 31;
    const int wave = threadIdx.x >> 5;
    const int wavesPerBlock = blockDim.x >> 5;
    const int colTiles = C_DIM / 16;             // 8
    const int rowStrips = (nRows + 31) >> 5;
    const int task = blockIdx.x * wavesPerBlock + wave;
    if (task >= rowStrips * colTiles) return;    // wave-uniform exit

    const int tm = task / colTiles;
    const int tn = task - tm * colTiles;
    const int m0 = tm * 32;
    const int n0 = tn * 16;

    if (m0 + 32 <= nRows) {                      // wave-uniform branch
        gemm_strip32<false>(H, Wnb, Wroot, bias, Ynb, Yroot, nRows, m0, n0, lane);
    } else {
        gemm_strip32<true>(H, Wnb, Wroot, bias, Ynb, Yroot, nRows, m0, n0, lane);
    }
}

// ---------------------------------------------------------------------------
extern "C" void kernel_launch(void* const* d_in, const int* in_sizes, int n_in,
                              void* d_out, int out_size, void* d_ws, size_t ws_size,
                              hipStream_t stream) {
    const float* x          = (const float*)d_in[0];
    const int*   pool_ei    = (const int*)d_in[1];
    const float* pool_attr  = (const float*)d_in[2];
    const int*   ei         = (const int*)d_in[3];
    const float* eattr      = (const float*)d_in[4];
    const float* W1_root    = (const float*)d_in[5];
    const float* W1_nb      = (const float*)d_in[6];
    const float* b1         = (const float*)d_in[7];
    const float* W2_root    = (const float*)d_in[8];
    const float* W2_nb      = (const float*)d_in[9];
    const float* b2         = (const float*)d_in[10];

    const long long E_pool = in_sizes[2];
    const long long E      = in_sizes[4];
    const int nC = out_size / C_DIM;             // 25000

    const size_t nodeElems = (size_t)nC * C_DIM;
    float* pooled = (float*)d_ws;
    float* y1nb   = pooled + nodeElems;
    float* h1     = y1nb + nodeElems;
    float* y2nb   = h1 + nodeElems;
    float* out    = (float*)d_out;

    // edge_index rows are flattened [src(0..E-1), dst(0..E-1)]
    const int* pool_src = pool_ei;
    const int* pool_dst = pool_ei + E_pool;
    const int* src      = ei;
    const int* dst      = ei + E;

    const int blk = 256;
    const int wavesPerBlock = blk / 32;
    const int rowStrips = (nC + 31) / 32;
    const int tasks = rowStrips * (C_DIM / 16);
    const int gemmGrid = (tasks + wavesPerBlock - 1) / wavesPerBlock;

    // 1) pooled = 0; pooled[dst] += w * x[src]
    hipMemsetAsync(pooled, 0, nodeElems * sizeof(float), stream);
    {
        long long total = E_pool * (C_DIM / 4);
        int grid = (int)((total + blk - 1) / blk);
        scatter_wadd4<<<grid, blk, 0, stream>>>(x, pool_src, pool_dst, pool_attr,
                                                pooled, E_pool);
    }

    // 2) layer 1 GEMMs: y1nb = pooled@W1_nb ; h1 = pooled@W1_root + b1
    gemm_dual_wmma<<<gemmGrid, blk, 0, stream>>>(pooled, W1_nb, W1_root, b1,
                                                 y1nb, h1, nC);

    // 3) h1[dst] += e * y1nb[src]
    {
        long long total = E * (C_DIM / 4);
        int grid = (int)((total + blk - 1) / blk);
        scatter_wadd4<<<grid, blk, 0, stream>>>(y1nb, src, dst, eattr, h1, E);
    }

    // 4) layer 2 GEMMs: y2nb = h1@W2_nb ; out = h1@W2_root + b2
    gemm_dual_wmma<<<gemmGrid, blk, 0, stream>>>(h1, W2_nb, W2_root, b2,
                                                 y2nb, out, nC);

    // 5) out[dst] += e * y2nb[src]
    {
        long long total = E * (C_DIM / 4);
        int grid = (int)((total + blk - 1) / blk);
        scatter_wadd4<<<grid, blk, 0, stream>>>(y2nb, src, dst, eattr, out, E);
    }
}